// PointNet_88210038325553
// MI455X (gfx1250) — compile-verified
//
#include <hip/hip_runtime.h>
#include <hip/hip_bf16.h>

typedef __attribute__((ext_vector_type(2))) float v2f;
typedef __attribute__((ext_vector_type(8))) float v8f;

#define C32 32
#define BPO 64                 // blocks per offset segment (27 segments: 26 nbr + center)
#define CONV_BLOCK 256         // 8 waves

// ---------------------------------------------------------------------------
// Generic zero-fill
// ---------------------------------------------------------------------------
__global__ __launch_bounds__(256) void zero_f32(float* __restrict__ p, long long n) {
    long long i = (long long)blockIdx.x * blockDim.x + threadIdx.x;
    long long stride = (long long)gridDim.x * blockDim.x;
    for (; i < n; i += stride) p[i] = 0.0f;
}

// ---------------------------------------------------------------------------
// Sparse conv: per-offset gather -> X[16,32] @ W[32,32] (fp32 WMMA) -> scatter-add
// seg 0..25 : neighbor offsets with kernel-map pairs (in_idx/out_idx, sentinel N)
// seg 26    : center tap (identity map, weight w_ctr)
// apply_bn  : fuse relu(x*scale[c] + shift[c]) on the gathered input (conv2)
// ---------------------------------------------------------------------------
__global__ __launch_bounds__(CONV_BLOCK)
void sparse_conv_wmma(const float* __restrict__ x,
                      const float* __restrict__ w_nb,     // [26,32,32]
                      const float* __restrict__ w_ctr,    // [32,32]
                      const long long* __restrict__ in_idx,   // [26,P]
                      const long long* __restrict__ out_idx,  // [26,P]
                      float* __restrict__ out,            // [N,32], pre-zeroed
                      int N, int P,
                      const float* __restrict__ bn_scale,
                      const float* __restrict__ bn_shift,
                      int apply_bn)
{
    const int lane = threadIdx.x & 31;
    const int wave = threadIdx.x >> 5;
    const int seg  = blockIdx.x / BPO;                 // 0..26
    const int wgrp = (blockIdx.x % BPO) * (CONV_BLOCK / 32) + wave;
    const int nwav = BPO * (CONV_BLOCK / 32);
    const int ln   = lane & 15;                        // row (A) / column (B,D) within tile
    const int half = lane >> 4;

    const bool center = (seg == 26);
    const float* __restrict__ W = center ? w_ctr : (w_nb + (size_t)seg * (C32 * C32));

    // B matrix resident in registers for the whole wave lifetime.
    // B k-step kk (rows 4kk..4kk+3), col half nh: VGPR0 = rows {4kk+2*half},
    // VGPR1 = rows {4kk+2*half+1}, column = nh*16 + ln   (ISA 32-bit B layout)
    v2f b[2][8];
#pragma unroll
    for (int nh = 0; nh < 2; ++nh)
#pragma unroll
        for (int kk = 0; kk < 8; ++kk) {
            const int r = 4 * kk + 2 * half;
            const int c = nh * 16 + ln;
            b[nh][kk].x = W[(size_t)r * C32 + c];
            b[nh][kk].y = W[(size_t)(r + 1) * C32 + c];
        }

    // Per-lane BN params for the channels this lane gathers (c0 = 4kk+2*half, c0+1)
    v2f bsc[8], bsh[8];
#pragma unroll
    for (int kk = 0; kk < 8; ++kk) { bsc[kk] = (v2f)(1.0f); bsh[kk] = (v2f)(0.0f); }
    if (apply_bn) {
#pragma unroll
        for (int kk = 0; kk < 8; ++kk) {
            const int c0 = 4 * kk + 2 * half;
            bsc[kk].x = bn_scale[c0];     bsc[kk].y = bn_scale[c0 + 1];
            bsh[kk].x = bn_shift[c0];     bsh[kk].y = bn_shift[c0 + 1];
        }
    }

    const int tiles = center ? (N + 15) / 16 : (P + 15) / 16;

    for (int t = wgrp; t < tiles; t += nwav) {
        const int p0 = t * 16;

        // Rows for this tile. Lanes L and L+16 compute identical values (both use ln).
        int inr = 0, outr = N;                 // defaults: dummy gather, skipped scatter
        if (center) {
            const int r = p0 + ln;
            if (r < N) { inr = r; outr = r; }
        } else {
            const int p = p0 + ln;
            if (p < P) {
                const long long i0 = in_idx [(size_t)seg * P + p];
                const long long o0 = out_idx[(size_t)seg * P + p];
                outr = (int)o0;
                inr  = (i0 < (long long)N) ? (int)i0 : 0;   // sentinel -> row 0 (discarded)
            }
        }

        // Gather A: lane holds row `ln`, channels {4kk+2*half, +1} per k-step.
        const float* __restrict__ xrow = x + (size_t)inr * C32;
        v2f a[8];
#pragma unroll
        for (int kk = 0; kk < 8; ++kk) {
            const int c0 = 4 * kk + 2 * half;
            v2f v = *(const v2f*)(xrow + c0);
            if (apply_bn) {
                v.x = fmaxf(v.x * bsc[kk].x + bsh[kk].x, 0.0f);
                v.y = fmaxf(v.y * bsc[kk].y + bsh[kk].y, 0.0f);
            }
            a[kk] = v;
        }

        // D[16,32] = A[16,32] @ W[32,32] as 2 x (8 chained K=4 fp32 WMMAs)
        v8f acc0 = {0.f,0.f,0.f,0.f,0.f,0.f,0.f,0.f};
        v8f acc1 = {0.f,0.f,0.f,0.f,0.f,0.f,0.f,0.f};
#pragma unroll
        for (int kk = 0; kk < 8; ++kk) {
            acc0 = __builtin_amdgcn_wmma_f32_16x16x4_f32(
                false, a[kk], false, b[0][kk], (short)0, acc0, false, false);
            acc1 = __builtin_amdgcn_wmma_f32_16x16x4_f32(
                false, a[kk], false, b[1][kk], (short)0, acc1, false, false);
        }

        // Scatter-add. D VGPR r holds row M = r + 8*half, column ln.
#pragma unroll
        for (int r = 0; r < 8; ++r) {
            const int M = r + 8 * half;
            const int orow = __shfl(outr, M);          // lane M (0..15) holds row M's index
            if (orow < N) {
                atomicAdd(&out[(size_t)orow * C32 + ln],       acc0[r]);
                atomicAdd(&out[(size_t)orow * C32 + 16 + ln],  acc1[r]);
            }
        }
    }
}

// ---------------------------------------------------------------------------
// Per-channel sum / sum-of-squares reduction (atomic accumulate into sums[64])
// ---------------------------------------------------------------------------
__global__ __launch_bounds__(256)
void reduce_stats(const float* __restrict__ h, int N, float* __restrict__ sums) {
    __shared__ float ssum[256];
    __shared__ float ssq[256];
    const int c = threadIdx.x & 31;
    const int g = threadIdx.x >> 5;   // 0..7
    float s = 0.0f, q = 0.0f;
    for (long long r = (long long)blockIdx.x * 8 + g; r < N; r += (long long)gridDim.x * 8) {
        const float v = h[r * C32 + c];
        s += v; q += v * v;
    }
    ssum[threadIdx.x] = s; ssq[threadIdx.x] = q;
    __syncthreads();
    if (threadIdx.x < 32) {
#pragma unroll
        for (int g2 = 1; g2 < 8; ++g2) { s += ssum[c + 32 * g2]; q += ssq[c + 32 * g2]; }
        atomicAdd(&sums[c], s);
        atomicAdd(&sums[32 + c], q);
    }
}

// ---------------------------------------------------------------------------
// BN fold: scale = g * rsqrt(var + eps), shift = b - mean*scale
// ---------------------------------------------------------------------------
__global__ void finalize_bn(const float* __restrict__ sums,
                            const float* __restrict__ gamma,
                            const float* __restrict__ beta,
                            int N, float* __restrict__ scale, float* __restrict__ shift) {
    const int c = threadIdx.x;
    if (c < C32) {
        const float inv_n = 1.0f / (float)N;
        const float mean = sums[c] * inv_n;
        const float var  = sums[32 + c] * inv_n - mean * mean;
        const float s    = gamma[c] * rsqrtf(var + 1e-5f);
        scale[c] = s;
        shift[c] = beta[c] - mean * s;
    }
}

// ---------------------------------------------------------------------------
// out = relu(BN2(h2) + feats), in place over d_out
// ---------------------------------------------------------------------------
__global__ __launch_bounds__(256)
void epilogue(float* __restrict__ out, const float* __restrict__ feats,
              const float* __restrict__ scale, const float* __restrict__ shift,
              long long total) {
    long long i = (long long)blockIdx.x * blockDim.x + threadIdx.x;
    const long long stride = (long long)gridDim.x * blockDim.x;
    for (; i < total; i += stride) {
        const int c = (int)(i & 31);
        out[i] = fmaxf(out[i] * scale[c] + shift[c] + feats[i], 0.0f);
    }
}

// ---------------------------------------------------------------------------
extern "C" void kernel_launch(void* const* d_in, const int* in_sizes, int n_in,
                              void* d_out, int out_size, void* d_ws, size_t ws_size,
                              hipStream_t stream) {
    const float*     feats  = (const float*)d_in[0];
    const float*     w1_nb  = (const float*)d_in[1];
    const float*     w1_ctr = (const float*)d_in[2];
    const float*     g1     = (const float*)d_in[3];
    const float*     b1     = (const float*)d_in[4];
    const float*     w2_nb  = (const float*)d_in[5];
    const float*     w2_ctr = (const float*)d_in[6];
    const float*     g2     = (const float*)d_in[7];
    const float*     b2     = (const float*)d_in[8];
    const long long* in_idx  = (const long long*)d_in[9];
    const long long* out_idx = (const long long*)d_in[10];

    const int N = in_sizes[0] / C32;
    const int P = in_sizes[9] / 26;
    float* out = (float*)d_out;

    // Workspace layout: h1[N*32] | sums1[64] | scale1[32] | shift1[32] | sums2[64] | scale2[32] | shift2[32]
    float* h1     = (float*)d_ws;
    float* stats  = h1 + (size_t)N * C32;
    float* sums1  = stats;
    float* scale1 = stats + 64;
    float* shift1 = stats + 96;
    float* sums2  = stats + 128;
    float* scale2 = stats + 192;
    float* shift2 = stats + 224;
    (void)ws_size; (void)n_in; (void)out_size;

    const long long elems = (long long)N * C32;
    const dim3 convGrid(27 * BPO);

    // 1) zero accumulators
    zero_f32<<<2048, 256, 0, stream>>>(h1, elems);
    zero_f32<<<2048, 256, 0, stream>>>(out, elems);
    zero_f32<<<1, 256, 0, stream>>>(stats, 256);

    // 2) conv1 -> h1
    sparse_conv_wmma<<<convGrid, CONV_BLOCK, 0, stream>>>(
        feats, w1_nb, w1_ctr, in_idx, out_idx, h1, N, P, nullptr, nullptr, 0);

    // 3) BN1 stats + fold
    reduce_stats<<<1024, 256, 0, stream>>>(h1, N, sums1);
    finalize_bn<<<1, 32, 0, stream>>>(sums1, g1, b1, N, scale1, shift1);

    // 4) conv2 with relu(BN1(.)) fused on the gather -> d_out
    sparse_conv_wmma<<<convGrid, CONV_BLOCK, 0, stream>>>(
        h1, w2_nb, w2_ctr, in_idx, out_idx, out, N, P, scale1, shift1, 1);

    // 5) BN2 stats + fold
    reduce_stats<<<1024, 256, 0, stream>>>(out, N, sums2);
    finalize_bn<<<1, 32, 0, stream>>>(sums2, g2, b2, N, scale2, shift2);

    // 6) out = relu(BN2(out) + feats)
    epilogue<<<4096, 256, 0, stream>>>(out, feats, scale2, shift2, elems);
}